// Triplet_128849019170
// MI455X (gfx1250) — compile-verified
//
#include <hip/hip_runtime.h>

typedef __attribute__((ext_vector_type(2))) float v2f;
typedef __attribute__((ext_vector_type(8))) float v8f;

#define TRIPLET_MARGIN 0.2f

// ---------------------------------------------------------------------------
// Stage 1: one block per row (D = 1024, 256 threads -> one float4 per tensor
// per thread). Computes diff[row] = sum_d (p-n)*(p+n-2a) = pos_dist - neg_dist.
// Wave-level reduction done with V_WMMA_F32_16X16X4_F32 against a ones
// B-matrix: D[m][n] = sum_k A[m][k], i.e. a ones-matmul is a lane reduction.
// ---------------------------------------------------------------------------
__global__ __launch_bounds__(256) void triplet_row_diff_kernel(
    const float* __restrict__ anc, const float* __restrict__ pos,
    const float* __restrict__ neg, float* __restrict__ diff)
{
    const int row  = blockIdx.x;
    const int t    = threadIdx.x;
    const size_t base = (size_t)row * 1024u + (size_t)t * 4u;

    const float4 av = *reinterpret_cast<const float4*>(anc + base);
    const float4 pv = *reinterpret_cast<const float4*>(pos + base);
    const float4 nv = *reinterpret_cast<const float4*>(neg + base);

    // (p-n) * (p+n-2a), accumulated over this thread's 4 elements
    float partial;
    partial  = (pv.x - nv.x) * fmaf(-2.0f, av.x, pv.x + nv.x);
    partial += (pv.y - nv.y) * fmaf(-2.0f, av.y, pv.y + nv.y);
    partial += (pv.z - nv.z) * fmaf(-2.0f, av.z, pv.z + nv.z);
    partial += (pv.w - nv.w) * fmaf(-2.0f, av.w, pv.w + nv.w);

    // --- wave32 reduction via WMMA ones-matmul (EXEC is all 1s here) ---
    // A (16x4 f32): VGPR0 = partial, VGPR1 = 0
    //   => A[m][0] = partial(lane m), A[m][2] = partial(lane m+16), rest 0.
    // B (4x16) = all ones  =>  D[m][n] = partial(lane m) + partial(lane m+16).
    v2f A; A.x = partial; A.y = 0.0f;
    v2f Bm; Bm.x = 1.0f; Bm.y = 1.0f;
    v8f C = {};
    C = __builtin_amdgcn_wmma_f32_16x16x4_f32(
            /*neg_a=*/false, A, /*neg_b=*/false, Bm,
            /*c_mod=*/(short)0, C, /*reuse_a=*/false, /*reuse_b=*/false);

    // C layout: lanes 0-15 hold rows 0..7 in VGPRs 0..7; lanes 16-31 rows 8..15.
    float s = C[0] + C[1] + C[2] + C[3] + C[4] + C[5] + C[6] + C[7];
    s += __shfl_xor(s, 16, 32);   // combine the two half-wave partial groups

    // --- cross-wave combine (8 waves) ---
    __shared__ float smem[8];
    const int wid = t >> 5;
    if ((t & 31) == 0) smem[wid] = s;
    __syncthreads();
    if (t == 0) {
        float tot = smem[0] + smem[1] + smem[2] + smem[3]
                  + smem[4] + smem[5] + smem[6] + smem[7];
        diff[row] = tot;
    }
}

// ---------------------------------------------------------------------------
// Stage 2: single block reduces B row-diffs into (loss, acc).
// Deterministic fixed-shape tree (no atomics).
// ---------------------------------------------------------------------------
__global__ __launch_bounds__(256) void triplet_finalize_kernel(
    const float* __restrict__ diff, float* __restrict__ out, int nrows)
{
    float relu_sum = 0.0f;
    float cnt      = 0.0f;
    for (int i = threadIdx.x; i < nrows; i += 256) {
        const float d = diff[i];                 // pos_dist - neg_dist
        relu_sum += fmaxf(d + TRIPLET_MARGIN, 0.0f);
        cnt      += (d > 0.0f) ? 1.0f : 0.0f;
    }

    #pragma unroll
    for (int off = 16; off > 0; off >>= 1) {
        relu_sum += __shfl_xor(relu_sum, off, 32);
        cnt      += __shfl_xor(cnt,      off, 32);
    }

    __shared__ float s_loss[8];
    __shared__ float s_cnt[8];
    const int wid = threadIdx.x >> 5;
    if ((threadIdx.x & 31) == 0) { s_loss[wid] = relu_sum; s_cnt[wid] = cnt; }
    __syncthreads();
    if (threadIdx.x == 0) {
        float L = 0.0f, A = 0.0f;
        #pragma unroll
        for (int i = 0; i < 8; ++i) { L += s_loss[i]; A += s_cnt[i]; }
        const float inv = 1.0f / (float)nrows;
        out[0] = L * inv;   // loss
        out[1] = A * inv;   // accuracy
    }
}

extern "C" void kernel_launch(void* const* d_in, const int* in_sizes, int n_in,
                              void* d_out, int out_size, void* d_ws, size_t ws_size,
                              hipStream_t stream) {
    (void)n_in; (void)out_size; (void)ws_size;
    const float* anc = (const float*)d_in[0];
    const float* pos = (const float*)d_in[1];
    const float* neg = (const float*)d_in[2];

    const int D = 1024;                 // feature dim (matches reference)
    const int B = in_sizes[0] / D;      // 16384 rows

    float* diff = (float*)d_ws;         // B floats of scratch (64 KB)

    triplet_row_diff_kernel<<<B, 256, 0, stream>>>(anc, pos, neg, diff);
    triplet_finalize_kernel<<<1, 256, 0, stream>>>(diff, (float*)d_out, B);
}